// SpectralConv2d_RC_40991167873033
// MI455X (gfx1250) — compile-verified
//
#include <hip/hip_runtime.h>
#include <hip/hip_bf16.h>
#include <math.h>

// ---------------------------------------------------------------------------
// FNO spectral conv (rfft2 -> mode-truncated complex GEMM -> irfft2) + CFT MLP
// All heavy stages use V_WMMA_F32_16X16X4_F32 (exact f32 math, wave32).
// Memory-bound problem (~8 GFLOP vs >=256MiB HBM): intermediates kept small
// and L2-resident; DFT matrices live in LDS in "paired" layout so each
// B-fragment is one ds_load_b64 per lane.
// ---------------------------------------------------------------------------

typedef float v2f __attribute__((ext_vector_type(2)));
typedef float v8f __attribute__((ext_vector_type(8)));

#define PI_F      3.14159265358979323846f
#define TWO_PI_F  6.28318530717958647692f

__device__ __forceinline__ v8f wmma4(v2f a, v2f b, v8f c) {
  // D = A(16x4 f32) x B(4x16 f32) + C(16x16 f32)
  return __builtin_amdgcn_wmma_f32_16x16x4_f32(false, a, false, b, (short)0, c,
                                               false, false);
}

__device__ __forceinline__ v2f ld2(const float* p) {
  return *reinterpret_cast<const v2f*>(p);
}

__device__ __forceinline__ void store8c(float* p, v8f v) {
  p[0] = v[0]; p[1] = v[1]; p[2] = v[2]; p[3] = v[3];
  p[4] = v[4]; p[5] = v[5]; p[6] = v[6]; p[7] = v[7];
}

// ===========================================================================
// Stage 1: W-DFT.  x[bc, h, w] -> Xw[bc, ky, h] (real & imag), ky = 0..15.
//   Xr = x . cos(2pi ky w/128),  Xi = -x . sin(2pi ky w/128)
// One wave per 16-h tile; D stored transposed ([ky][h]) for stage 2 A-loads.
// ===========================================================================
__global__ __launch_bounds__(256) void k_dftW(const float* __restrict__ x,
                                              float* __restrict__ XwR,
                                              float* __restrict__ XwI) {
  __shared__ float tC[2048];   // paired: [pair p][n=ky][2] -> B[2p,n],B[2p+1,n]
  __shared__ float tS[2048];
  for (int i = threadIdx.x; i < 2048; i += 256) {
    int p = i >> 5, rem = i & 31;
    int n = rem >> 1, half = rem & 1;
    int k = p * 2 + half;                                  // k = w index
    float ang = (TWO_PI_F / 128.0f) * (float)(k * n);
    tC[i] = cosf(ang);
    tS[i] = -sinf(ang);
  }
  __syncthreads();
  int lane = threadIdx.x & 31, wave = threadIdx.x >> 5;
  int tile = blockIdx.x * 8 + wave;                        // 0..16383
  int bc = tile >> 3;
  int h0 = (tile & 7) << 4;
  int nl = lane & 15, phl = lane >> 4;
  const float* aBase = x + (size_t)bc * 16384 +
                       (size_t)(h0 + nl) * 128 + (phl << 1);
  v8f accC = {0, 0, 0, 0, 0, 0, 0, 0};
  v8f accS = {0, 0, 0, 0, 0, 0, 0, 0};
  for (int k0 = 0; k0 < 128; k0 += 4) {
    v2f a = ld2(aBase + k0);
    int p = (k0 >> 1) + phl;
    v2f bc2 = ld2(&tC[(p << 5) + (nl << 1)]);
    v2f bs2 = ld2(&tS[(p << 5) + (nl << 1)]);
    accC = wmma4(a, bc2, accC);
    accS = wmma4(a, bs2, accS);
  }
  int mo = phl << 3;
  float* dR = XwR + (size_t)bc * 2048 + (size_t)nl * 128 + h0 + mo;
  float* dI = XwI + (size_t)bc * 2048 + (size_t)nl * 128 + h0 + mo;
  store8c(dR, accC);
  store8c(dI, accS);
}

// ===========================================================================
// Stage 2: H-DFT on 32 selected kx rows ({0..15} and {112..127}).
//   G = Xw * e^{-2pi i kx h/128}:  Gr = Xr C + Xi S,  Gi = Xi C - Xr S
// Input Xw[bc, ky, h]; one wave per bc (16 ky rows). Output G[kxsel,ky,b,c].
// ===========================================================================
__global__ __launch_bounds__(256) void k_dftH(const float* __restrict__ XwR,
                                              const float* __restrict__ XwI,
                                              float* __restrict__ GR,
                                              float* __restrict__ GI) {
  __shared__ float tCl[2048], tSl[2048], tCh[2048], tSh[2048];
  for (int i = threadIdx.x; i < 2048; i += 256) {
    int p = i >> 5, rem = i & 31;
    int n = rem >> 1, half = rem & 1;
    int h = p * 2 + half;
    float al = (TWO_PI_F / 128.0f) * (float)(n * h);           // kx = n
    float ah = (TWO_PI_F / 128.0f) * (float)((112 + n) * h);   // kx = 112+n
    tCl[i] = cosf(al); tSl[i] = sinf(al);
    tCh[i] = cosf(ah); tSh[i] = sinf(ah);
  }
  __syncthreads();
  int lane = threadIdx.x & 31, wave = threadIdx.x >> 5;
  int bc = blockIdx.x * 8 + wave;                          // 0..2047
  int nl = lane & 15, phl = lane >> 4;
  const float* aR = XwR + (size_t)bc * 2048 + (size_t)nl * 128 + (phl << 1);
  const float* aI = XwI + (size_t)bc * 2048 + (size_t)nl * 128 + (phl << 1);
  v8f rlo = {0, 0, 0, 0, 0, 0, 0, 0}, ilo = rlo, rhi = rlo, ihi = rlo;
  for (int k0 = 0; k0 < 128; k0 += 4) {
    v2f ar = ld2(aR + k0);
    v2f ai = ld2(aI + k0);
    int q = (((k0 >> 1) + phl) << 5) + (nl << 1);
    v2f cl = ld2(&tCl[q]), sl = ld2(&tSl[q]);
    v2f ch = ld2(&tCh[q]), sh = ld2(&tSh[q]);
    rlo = wmma4(ar, cl, rlo);
    rlo = wmma4(ai, sl, rlo);
    ilo = wmma4(ai, cl, ilo);
    ilo = wmma4(ar, -sl, ilo);
    rhi = wmma4(ar, ch, rhi);
    rhi = wmma4(ai, sh, rhi);
    ihi = wmma4(ai, ch, ihi);
    ihi = wmma4(ar, -sh, ihi);
  }
  int b = bc >> 6, c = bc & 63;
  int mo = phl << 3;
#pragma unroll
  for (int j = 0; j < 8; ++j) {
    int M = mo + j;                                         // ky
    size_t alo = ((size_t)(nl * 16 + M) * 32 + b) * 64 + c;
    size_t ahi = ((size_t)((nl + 16) * 16 + M) * 32 + b) * 64 + c;
    GR[alo] = rlo[j]; GI[alo] = ilo[j];
    GR[ahi] = rhi[j]; GI[ahi] = ihi[j];
  }
}

// ===========================================================================
// Stage 3: spectral multiply per mode: out[b,o] = sum_c G[b,c] * W[c,o]
// (complex). One block per mode (kxsel,ky); wave -> (b-tile, o-tile).
// Output F[b, o, ky, kxsel].
// ===========================================================================
__global__ __launch_bounds__(256) void k_specmul(const float* __restrict__ GR,
                                                 const float* __restrict__ GI,
                                                 const float* __restrict__ w1,
                                                 const float* __restrict__ w2,
                                                 float* __restrict__ FR,
                                                 float* __restrict__ FI) {
  int kxsel = blockIdx.x >> 4;
  int ky = blockIdx.x & 15;
  int lane = threadIdx.x & 31, wave = threadIdx.x >> 5;
  int mt = wave & 1, nt = wave >> 1;
  int nl = lane & 15, phl = lane >> 4;
  const float* Wp = (kxsel < 16) ? w1 : w2;
  int kxw = kxsel & 15;
  size_t modeoff = (size_t)(kxsel * 16 + ky) * 2048;       // G slice [b][c]
  const float* aR = GR + modeoff + (size_t)(mt * 16 + nl) * 64 + (phl << 1);
  const float* aI = GI + modeoff + (size_t)(mt * 16 + nl) * 64 + (phl << 1);
  int o = nt * 16 + nl;
  v8f accr = {0, 0, 0, 0, 0, 0, 0, 0}, acci = accr;
  for (int k0 = 0; k0 < 64; k0 += 4) {
    v2f ar = ld2(aR + k0);
    v2f ai = ld2(aI + k0);
    int c0 = k0 + (phl << 1);
    size_t i0 = (((size_t)c0 * 64 + o) * 256 + (size_t)(kxw * 16 + ky)) * 2;
    size_t i1 = i0 + 32768;                                // c0+1
    v2f br = {Wp[i0], Wp[i1]};
    v2f bi = {Wp[i0 + 1], Wp[i1 + 1]};
    accr = wmma4(ar, br, accr);     // Gr*Wr
    accr = wmma4(ai, -bi, accr);    // - Gi*Wi
    acci = wmma4(ar, bi, acci);     // Gr*Wi
    acci = wmma4(ai, br, acci);     // + Gi*Wr
  }
  int mo = phl << 3;
#pragma unroll
  for (int j = 0; j < 8; ++j) {
    int b = mt * 16 + mo + j;
    size_t addr = (((size_t)b * 64 + o) * 16 + ky) * 32 + kxsel;
    FR[addr] = accr[j];
    FI[addr] = acci[j];
  }
}

// ===========================================================================
// Stage 4: inverse DFT along H (complex, e^{+2pi i kx h/128}, no 1/H yet):
//   Yr = Fr C - Fi S,  Yi = Fr S + Fi C
// Input F[bo, ky, kxsel]; one wave per bo. Output Y[bo, h, ky] (transposed).
// ===========================================================================
__global__ __launch_bounds__(256) void k_idftH(const float* __restrict__ FR,
                                               const float* __restrict__ FI,
                                               float* __restrict__ YR,
                                               float* __restrict__ YI) {
  __shared__ float tC[4096], tS[4096];                     // [pair(16)][h(128)][2]
  for (int i = threadIdx.x; i < 4096; i += 256) {
    int p = i >> 8, rem = i & 255;
    int n = rem >> 1, half = rem & 1;
    int k = p * 2 + half;                                  // kxsel 0..31
    int kx = (k < 16) ? k : (96 + k);                      // 112..127
    float ang = (TWO_PI_F / 128.0f) * (float)(kx * n);
    tC[i] = cosf(ang);
    tS[i] = sinf(ang);
  }
  __syncthreads();
  int lane = threadIdx.x & 31, wave = threadIdx.x >> 5;
  int bo = blockIdx.x * 8 + wave;                          // 0..2047
  int nl = lane & 15, phl = lane >> 4;
  const float* aR = FR + (size_t)bo * 512 + (size_t)nl * 32 + (phl << 1);
  const float* aI = FI + (size_t)bo * 512 + (size_t)nl * 32 + (phl << 1);
  v8f zero = {0, 0, 0, 0, 0, 0, 0, 0};
  v8f accr[8], acci[8];
#pragma unroll
  for (int t = 0; t < 8; ++t) { accr[t] = zero; acci[t] = zero; }
  for (int k0 = 0; k0 < 32; k0 += 4) {
    v2f ar = ld2(aR + k0);
    v2f ai = ld2(aI + k0);
    int p = (k0 >> 1) + phl;
#pragma unroll
    for (int ht = 0; ht < 8; ++ht) {
      int q = (p << 8) + (((ht << 4) + nl) << 1);
      v2f bc2 = ld2(&tC[q]);
      v2f bs2 = ld2(&tS[q]);
      accr[ht] = wmma4(ar, bc2, accr[ht]);
      accr[ht] = wmma4(ai, -bs2, accr[ht]);
      acci[ht] = wmma4(ar, bs2, acci[ht]);
      acci[ht] = wmma4(ai, bc2, acci[ht]);
    }
  }
  int mo = phl << 3;
#pragma unroll
  for (int ht = 0; ht < 8; ++ht) {
    int h = ht * 16 + nl;
    float* pR = YR + ((size_t)bo * 128 + h) * 16 + mo;
    float* pI = YI + ((size_t)bo * 128 + h) * 16 + mo;
    store8c(pR, accr[ht]);
    store8c(pI, acci[ht]);
  }
}

// ===========================================================================
// Stage 5: irfft along W with Hermitian weights (ky=0 weight 1, else 2),
// scaled by 1/(H*W); adds corr[b,o]; coalesced final stores.
//   out = sum_ky Yr * (a cos)/16384 + Yi * (-a sin)/16384 + corr
// ===========================================================================
__global__ __launch_bounds__(256) void k_irfftW(const float* __restrict__ YR,
                                                const float* __restrict__ YI,
                                                const float* __restrict__ corr,
                                                float* __restrict__ out) {
  __shared__ float tBr[2048], tBi[2048];                   // [pair(8)][w(128)][2]
  for (int i = threadIdx.x; i < 2048; i += 256) {
    int p = i >> 8, rem = i & 255;
    int n = rem >> 1, half = rem & 1;
    int k = p * 2 + half;                                  // ky
    float alpha = (k == 0) ? 1.0f : 2.0f;
    float s = alpha / 16384.0f;
    float ang = (TWO_PI_F / 128.0f) * (float)(k * n);
    tBr[i] = s * cosf(ang);
    tBi[i] = -s * sinf(ang);
  }
  __syncthreads();
  int lane = threadIdx.x & 31, wave = threadIdx.x >> 5;
  int tile = blockIdx.x * 8 + wave;                        // 0..16383
  int bo = tile >> 3;
  int h0 = (tile & 7) << 4;
  int nl = lane & 15, phl = lane >> 4;
  const float* aR = YR + ((size_t)bo * 128 + h0 + nl) * 16 + (phl << 1);
  const float* aI = YI + ((size_t)bo * 128 + h0 + nl) * 16 + (phl << 1);
  v8f zero = {0, 0, 0, 0, 0, 0, 0, 0};
  v8f acc[8];
#pragma unroll
  for (int t = 0; t < 8; ++t) acc[t] = zero;
  for (int k0 = 0; k0 < 16; k0 += 4) {
    v2f ar = ld2(aR + k0);
    v2f ai = ld2(aI + k0);
    int p = (k0 >> 1) + phl;
#pragma unroll
    for (int wt = 0; wt < 8; ++wt) {
      int q = (p << 8) + (((wt << 4) + nl) << 1);
      v2f br = ld2(&tBr[q]);
      v2f bi = ld2(&tBi[q]);
      acc[wt] = wmma4(ar, br, acc[wt]);
      acc[wt] = wmma4(ai, bi, acc[wt]);
    }
  }
  float cv = corr[bo];
  int mo = phl << 3;
#pragma unroll
  for (int wt = 0; wt < 8; ++wt) {
#pragma unroll
    for (int j = 0; j < 8; ++j) {
      size_t addr = (size_t)bo * 16384 + (size_t)(h0 + mo + j) * 128 +
                    (size_t)(wt * 16 + nl);
      out[addr] = acc[wt][j] + cv;
    }
  }
}

// ===========================================================================
// CFT correction path (tiny): Chebyshev-segment quadrature transform.
// ===========================================================================
__device__ __forceinline__ float wred32(float v) {
  v += __shfl_xor(v, 16, 32);
  v += __shfl_xor(v, 8, 32);
  v += __shfl_xor(v, 4, 32);
  v += __shfl_xor(v, 2, 32);
  v += __shfl_xor(v, 1, 32);
  return v;
}

// Build per-(l,m) interpolation indices/weights and effective quadrature
// weights ER/EI[(l,m)][f] = Wr[m,f]*cos(ph) -/+ Wi[m,f]*sin(ph) combos.
__device__ __forceinline__ void build_cft_tables(const float* fpts,
                                                 int* sLi, int* sRi,
                                                 float* sWl, float* sWr,
                                                 float (*sER)[4],
                                                 float (*sEI)[4]) {
  int i = threadIdx.x;
  if (i < 32) {
    int l = i >> 3, m = i & 7;
    float node = -cosf((2.0f * (float)m + 1.0f) * PI_F / 16.0f);
    float tseg = 0.25f * (float)l + 0.125f * (node + 1.0f);
    // searchsorted(left) on t[j] = j/127
    int ri = 0;
    while (ri < 127 && (float)ri * (1.0f / 127.0f) < tseg) ++ri;
    int li = ri - 1;
    float tl = (float)li * (1.0f / 127.0f);
    float tr = (float)ri * (1.0f / 127.0f);
    float den = ((tr - tl) < 1e-8f) ? 1.0f : (tr - tl);
    float wr = (tseg - tl) / den;
    sLi[i] = li; sRi[i] = ri;
    sWl[i] = 1.0f - wr; sWr[i] = wr;
    for (int f = 0; f < 4; ++f) {
      float fp = fpts[f];
      float wprime = fp * TWO_PI_F * 0.125f;
      float Wr_ = 0.0f, Wi_ = 0.0f;
      for (int mp = 0; mp < 8; ++mp) {
        float nodep = -cosf((2.0f * (float)mp + 1.0f) * PI_F / 16.0f);
        float T = cosf((float)m * acosf(nodep));
        float ang = nodep * wprime;
        Wr_ += T * cosf(ang);
        Wi_ += T * (-sinf(ang));
      }
      Wr_ *= 0.125f; Wi_ *= 0.125f;
      float phase = -TWO_PI_F * (0.25f * (float)l) * fp;
      float cr = cosf(phase), ci = sinf(phase);
      sER[i][f] = Wr_ * cr - Wi_ * ci;
      sEI[i][f] = Wr_ * ci + Wi_ * cr;
    }
  }
  __syncthreads();
}

// CFT along W: one wave per (b,c,h) row; lane = (l,m). Out r1/i1[row][f2].
__global__ __launch_bounds__(256) void k_cft_w(const float* __restrict__ x,
                                               float* __restrict__ r1,
                                               float* __restrict__ i1) {
  __shared__ int sLi[32], sRi[32];
  __shared__ float sWl[32], sWr[32];
  __shared__ float sER[32][4], sEI[32][4];
  const float fpts[4] = {0.0f, 1.0f, 2.0f, 3.0f};
  build_cft_tables(fpts, sLi, sRi, sWl, sWr, sER, sEI);
  int lane = threadIdx.x & 31, wave = threadIdx.x >> 5;
  size_t row = (size_t)blockIdx.x * 8 + wave;              // 0..262143
  const float* xr = x + row * 128;
  float v = sWl[lane] * xr[sLi[lane]] + sWr[lane] * xr[sRi[lane]];
  float rv[4], iv[4];
#pragma unroll
  for (int f = 0; f < 4; ++f) {
    rv[f] = wred32(v * sER[lane][f]);
    iv[f] = wred32(v * sEI[lane][f]);
  }
  if (lane == 0) {
    float* pr = r1 + row * 4;
    float* pi = i1 + row * 4;
    pr[0] = rv[0]; pr[1] = rv[1]; pr[2] = rv[2]; pr[3] = rv[3];
    pi[0] = iv[0]; pi[1] = iv[1]; pi[2] = iv[2]; pi[3] = iv[3];
  }
}

// CFT along H on (r1 + i*i1): one wave per (b,c,f2); frequencies [0,1,-2,-1].
// Writes flat[b, (c*4+f1)*4+f2, {re,im}].
__global__ __launch_bounds__(256) void k_cft_h(const float* __restrict__ r1,
                                               const float* __restrict__ i1,
                                               float* __restrict__ flat) {
  __shared__ int sLi[32], sRi[32];
  __shared__ float sWl[32], sWr[32];
  __shared__ float sER[32][4], sEI[32][4];
  const float fpts[4] = {0.0f, 1.0f, -2.0f, -1.0f};
  build_cft_tables(fpts, sLi, sRi, sWl, sWr, sER, sEI);
  int lane = threadIdx.x & 31, wave = threadIdx.x >> 5;
  int row = blockIdx.x * 8 + wave;                         // 0..8191
  int bc = row >> 2, f2 = row & 3;
  const float* rp = r1 + (size_t)bc * 512 + f2;            // stride 4 over h
  const float* ip = i1 + (size_t)bc * 512 + f2;
  float segR = sWl[lane] * rp[sLi[lane] * 4] + sWr[lane] * rp[sRi[lane] * 4];
  float segI = sWl[lane] * ip[sLi[lane] * 4] + sWr[lane] * ip[sRi[lane] * 4];
  float crv[4], civ[4];
#pragma unroll
  for (int f = 0; f < 4; ++f) {
    // real = rr - ii ; imag = ri + ir
    crv[f] = wred32(segR * sER[lane][f] - segI * sEI[lane][f]);
    civ[f] = wred32(segR * sEI[lane][f] + segI * sER[lane][f]);
  }
  if (lane == 0) {
    int b = bc >> 6, c = bc & 63;
    float* pb = flat + (size_t)b * 2048;
#pragma unroll
    for (int f = 0; f < 4; ++f) {
      size_t idx = (size_t)((c * 4 + f) * 4 + f2) * 2;
      pb[idx] = crv[f];
      pb[idx + 1] = civ[f];
    }
  }
}

// MLP: corr[b,o] = gelu(flat @ W1 + b1) @ W2 + b2. One block per batch row.
__global__ __launch_bounds__(256) void k_mlp(const float* __restrict__ flat,
                                             const float* __restrict__ W1,
                                             const float* __restrict__ b1,
                                             const float* __restrict__ W2,
                                             const float* __restrict__ b2,
                                             float* __restrict__ corr) {
  __shared__ float sIn[2048];
  __shared__ float sH[256];
  int b = blockIdx.x;
  for (int i = threadIdx.x; i < 2048; i += 256) sIn[i] = flat[(size_t)b * 2048 + i];
  __syncthreads();
  int j = threadIdx.x;
  float acc = b1[j];
  for (int i = 0; i < 2048; ++i) acc = fmaf(sIn[i], W1[(size_t)i * 256 + j], acc);
  // exact GELU: 0.5*x*(1+erf(x/sqrt(2)))
  sH[j] = 0.5f * acc * (1.0f + erff(acc * 0.70710678118654752f));
  __syncthreads();
  if (j < 64) {
    float a2 = b2[j];
    for (int t = 0; t < 256; ++t) a2 = fmaf(sH[t], W2[(size_t)t * 64 + j], a2);
    corr[(size_t)b * 64 + j] = a2;
  }
}

// ===========================================================================
// Launch: workspace regions reused across phases (~50.3 MB total):
//   [0,4M)   XwR  -> later YR          [4M,8M)  XwI -> later YI
//   [8M,9M)  r1   -> later GR          [9M,10M) i1  -> later GI
//   [10M,11M) flat -> later FR         [11M,12M) FI
//   [12M, +2048) corr
// ===========================================================================
extern "C" void kernel_launch(void* const* d_in, const int* in_sizes, int n_in,
                              void* d_out, int out_size, void* d_ws,
                              size_t ws_size, hipStream_t stream) {
  (void)in_sizes; (void)n_in; (void)out_size; (void)ws_size;
  const float* x   = (const float*)d_in[0];
  const float* w1  = (const float*)d_in[1];
  const float* w2  = (const float*)d_in[2];
  const float* l1w = (const float*)d_in[3];
  const float* l1b = (const float*)d_in[4];
  const float* l2w = (const float*)d_in[5];
  const float* l2b = (const float*)d_in[6];
  float* out = (float*)d_out;
  float* ws = (float*)d_ws;

  float* XwR  = ws + 0;         // 4,194,304 floats (aliases YR)
  float* XwI  = ws + 4194304;   // (aliases YI)
  float* GR   = ws + 8388608;   // 1,048,576 (aliases r1)
  float* GI   = ws + 9437184;   // (aliases i1)
  float* FR   = ws + 10485760;  // 1,048,576 (aliases flat)
  float* FI   = ws + 11534336;
  float* corr = ws + 12582912;  // 2048
  float* r1 = GR;
  float* i1 = GI;
  float* flat = FR;
  float* YR = XwR;
  float* YI = XwI;

  // Correction path first (its buffers are reused by later spectral stages).
  k_cft_w<<<dim3(32768), dim3(256), 0, stream>>>(x, r1, i1);
  k_cft_h<<<dim3(1024), dim3(256), 0, stream>>>(r1, i1, flat);
  k_mlp<<<dim3(32), dim3(256), 0, stream>>>(flat, l1w, l1b, l2w, l2b, corr);

  // Spectral pipeline (all WMMA f32 16x16x4).
  k_dftW<<<dim3(2048), dim3(256), 0, stream>>>(x, XwR, XwI);
  k_dftH<<<dim3(256), dim3(256), 0, stream>>>(XwR, XwI, GR, GI);
  k_specmul<<<dim3(512), dim3(256), 0, stream>>>(GR, GI, w1, w2, FR, FI);
  k_idftH<<<dim3(256), dim3(256), 0, stream>>>(FR, FI, YR, YI);
  k_irfftW<<<dim3(2048), dim3(256), 0, stream>>>(YR, YI, corr, out);
}